// MWT_CZ1d_31473520345372
// MI455X (gfx1250) — compile-verified
//
#include <hip/hip_runtime.h>
#include <hip/hip_bf16.h>
#include <math.h>
#include <stdint.h>

typedef float v2f __attribute__((ext_vector_type(2)));
typedef float v8f __attribute__((ext_vector_type(8)));

// ---------------------------------------------------------------------------
// D = A(16x4 f32) * B(4x16 f32) + C(16x16 f32)  -- CDNA5 V_WMMA_F32_16X16X4_F32
// ---------------------------------------------------------------------------
static __device__ __forceinline__ v8f wmma4(v2f a, v2f b, v8f c) {
  return __builtin_amdgcn_wmma_f32_16x16x4_f32(false, a, false, b, (short)0, c,
                                               false, false);
}

// ---------------------------------------------------------------------------
// CDNA5 async global->LDS copy (ASYNCcnt-tracked), per-lane 4B.
// ---------------------------------------------------------------------------
static __device__ __forceinline__ void async_copy_b32(void* lds_ptr,
                                                      const void* gptr) {
  uint32_t lds_off = (uint32_t)(uintptr_t)lds_ptr;
  asm volatile("global_load_async_to_lds_b32 %0, %1, off" ::"v"(lds_off),
               "v"(gptr)
               : "memory");
}
static __device__ __forceinline__ void async_wait0() {
  asm volatile("s_wait_asynccnt 0" ::: "memory");
}

// ---------------------------------------------------------------------------
// Twiddle table: tw[p] = (cos, sin)(2*pi*p/4096). One table serves all levels:
// exp(-2*pi*i*j*m/n) = (cos,-sin)[ (j*(4096/n)*m) & 4095 ]
// ---------------------------------------------------------------------------
__global__ void tw_init(float* __restrict__ twc, float* __restrict__ tws) {
  int p = blockIdx.x * 256 + threadIdx.x;
  if (p < 4096) {
    float th = (6.2831853071795864769f / 4096.0f) * (float)p;
    twc[p] = cosf(th);
    tws[p] = sinf(th);
  }
}

// ---------------------------------------------------------------------------
// Wavelet analysis step:
//   xa[b,j,c,t] = t<8 ? x[b,2j,c,t] : x[b,2j+1,c,t-8]   (t in 0..15)
//   xs = xa @ ec_s (16x8),  xd = xa @ ec_d (16x8)
// ---------------------------------------------------------------------------
__global__ void wavelet_fwd(const float* __restrict__ x,
                            const float* __restrict__ ec_s,
                            const float* __restrict__ ec_d,
                            float* __restrict__ xs, float* __restrict__ xd,
                            int n) {
  int idx = blockIdx.x * 256 + threadIdx.x;
  if (idx >= (n << 11)) return;
  int k = idx & 7;
  int c = (idx >> 3) & 15;
  int j = (idx >> 7) % n;
  int b = (idx >> 7) / n;
  const float* xe = x + ((size_t)((b * (n << 1)) + (j << 1)) * 16 + c) * 8;
  const float* xo = xe + 128;
  float as = 0.f, ad = 0.f;
#pragma unroll
  for (int t = 0; t < 8; ++t) {
    float ve = xe[t], vo = xo[t];
    as = fmaf(ve, ec_s[t * 8 + k], as);
    as = fmaf(vo, ec_s[(t + 8) * 8 + k], as);
    ad = fmaf(ve, ec_d[t * 8 + k], ad);
    ad = fmaf(vo, ec_d[(t + 8) * 8 + k], ad);
  }
  xs[idx] = as;
  xd[idx] = ad;
}

// ---------------------------------------------------------------------------
// Truncated forward rDFT (ortho) as WMMA GEMM:
//   FT[m,ch] = (1/sqrt(n)) sum_j  Tw[m,j] * v[j,ch]      (cos and sin planes)
// M = 16 modes/block.x, N = 128 ch (8 waves x 16), K = j in steps of 4.
// Twiddle table (32KB) async-staged to LDS; A-fragments gathered per lane.
// Hot path (n >= 4) has a guard-free inner loop; n in {1,2} uses branchless
// clamped-address + value-mask loads so EXEC stays all-ones for WMMA.
// grid ((l+15)/16, 16), block 256.
// ---------------------------------------------------------------------------
__global__ void __launch_bounds__(256) dft_fwd_wmma(
    const float* __restrict__ v, float* __restrict__ fr, float* __restrict__ fi,
    const float* __restrict__ twc, const float* __restrict__ tws, int n) {
  __shared__ float s_twc[4096];
  __shared__ float s_tws[4096];
  const int tid = threadIdx.x;
  for (int i = tid; i < 4096; i += 256) {
    async_copy_b32(&s_twc[i], &twc[i]);
    async_copy_b32(&s_tws[i], &tws[i]);
  }
  async_wait0();
  __syncthreads();

  const int m0 = blockIdx.x << 4;
  const int b = blockIdx.y;
  const unsigned s = 4096u / (unsigned)n;
  const int lane = tid & 31;
  const int ch0 = (tid >> 5) << 4;  // wave -> N tile
  const int bm = lane & 15;         // A: M offset / B: N offset
  const int kh = (lane >> 4) << 1;  // lane-half K pair offset
  const unsigned mA = (unsigned)(m0 + bm);
  const float* vb = v + ((size_t)b * (size_t)n) * 128 + (ch0 + bm);

  v8f ac = {0.f, 0.f, 0.f, 0.f, 0.f, 0.f, 0.f, 0.f};
  v8f as_ = ac;
  if (n >= 4) {  // uniform branch; all lanes of every K-step are in-range
    for (int j0 = 0; j0 < n; j0 += 4) {
      int ja = j0 + kh;
      unsigned p0 = ((unsigned)ja * s * mA) & 4095u;
      unsigned p1 = ((unsigned)(ja + 1) * s * mA) & 4095u;
      v2f a_c = {s_twc[p0], s_twc[p1]};
      v2f a_s = {s_tws[p0], s_tws[p1]};
      v2f b_v = {vb[(size_t)ja << 7], vb[(size_t)(ja + 1) << 7]};
      ac = wmma4(a_c, b_v, ac);
      as_ = wmma4(a_s, b_v, as_);
    }
  } else {  // n in {1,2}: single K-step, mask K-rows >= n without branching
    int ja = kh;
    unsigned p0 = ((unsigned)ja * s * mA) & 4095u;
    unsigned p1 = ((unsigned)(ja + 1) * s * mA) & 4095u;
    v2f a_c = {s_twc[p0], s_twc[p1]};
    v2f a_s = {s_tws[p0], s_tws[p1]};
    int ja0 = (ja < n) ? ja : 0;
    int ja1 = ((ja + 1) < n) ? (ja + 1) : 0;
    float m0v = (ja < n) ? 1.f : 0.f;
    float m1v = ((ja + 1) < n) ? 1.f : 0.f;
    v2f b_v = {vb[(size_t)ja0 << 7] * m0v, vb[(size_t)ja1 << 7] * m1v};
    ac = wmma4(a_c, b_v, ac);
    as_ = wmma4(a_s, b_v, as_);
  }
  const float sc = rsqrtf((float)n);
  const int mhi = (lane >> 4) << 3;
#pragma unroll
  for (int r = 0; r < 8; ++r) {
    int mm = m0 + r + mhi;  // <= 63 by construction
    size_t g = ((size_t)((b << 6) + mm) << 7) + (ch0 + bm);
    fr[g] = ac[r] * sc;
    fi[g] = -as_[r] * sc;
  }
}

// ---------------------------------------------------------------------------
// Truncated inverse rDFT (ortho) as WMMA GEMM:
//   x[j,ch] = (1/sqrt(n)) sum_m [ w_m cos(th) * gr[m,ch] - w_m sin(th) * gi[m,ch] ]
// w_m = 1 for DC/Nyquist, 2 otherwise, 0 for m >= l (mode mask folded into A).
// M = 16 time positions/block.x, N = 128 ch, K = modes in steps of 4.
// grid ((n+15)/16, 16), block 256.
// ---------------------------------------------------------------------------
__global__ void __launch_bounds__(256) dft_inv_wmma(
    const float* __restrict__ gr, const float* __restrict__ gi,
    float* __restrict__ out, const float* __restrict__ twc,
    const float* __restrict__ tws, int n, int l) {
  const int tid = threadIdx.x;
  const int j0 = blockIdx.x << 4;
  const int b = blockIdx.y;
  const unsigned s = 4096u / (unsigned)n;
  const int lane = tid & 31;
  const int ch0 = (tid >> 5) << 4;
  const int bm = lane & 15;
  const int kh = (lane >> 4) << 1;
  const unsigned jA = (unsigned)(j0 + bm);
  const int nyqm = n >> 1;
  const float* grb = gr + ((size_t)(b << 6) << 7) + (ch0 + bm);
  const float* gib = gi + ((size_t)(b << 6) << 7) + (ch0 + bm);

  v8f acc = {0.f, 0.f, 0.f, 0.f, 0.f, 0.f, 0.f, 0.f};
  for (int m0k = 0; m0k < l; m0k += 4) {
    int ma0 = m0k + kh;
    int ma1 = ma0 + 1;
    float w0 = (ma0 >= l) ? 0.f : ((ma0 == 0 || ma0 == nyqm) ? 1.f : 2.f);
    float w1 = (ma1 >= l) ? 0.f : ((ma1 == 0 || ma1 == nyqm) ? 1.f : 2.f);
    unsigned p0 = (jA * s * (unsigned)ma0) & 4095u;
    unsigned p1 = (jA * s * (unsigned)ma1) & 4095u;
    v2f a_c = {w0 * twc[p0], w1 * twc[p1]};
    v2f a_s = {-w0 * tws[p0], -w1 * tws[p1]};
    int mb0 = ma0 & 63, mb1 = ma1 & 63;  // stay in 64-slot stride; w=0 masks
    v2f b_r = {grb[(size_t)mb0 << 7], grb[(size_t)mb1 << 7]};
    v2f b_i = {gib[(size_t)mb0 << 7], gib[(size_t)mb1 << 7]};
    acc = wmma4(a_c, b_r, acc);
    acc = wmma4(a_s, b_i, acc);
  }
  const float sc = rsqrtf((float)n);
  const int jhi = (lane >> 4) << 3;
#pragma unroll
  for (int r = 0; r < 8; ++r) {
    int jj = j0 + r + jhi;
    if (jj < n)
      out[(((size_t)b * n + jj) << 7) + (ch0 + bm)] = acc[r] * sc;
  }
}

// ---------------------------------------------------------------------------
// Per-mode complex GEMMs with f32 WMMA. One workgroup per mode m:
//   Ud_hat[b,o] = sum_i ftd[b,i]*(Awr+iAwi)[i,o,m] + ftx[b,i]*(Bwr+iBwi)[i,o,m]
//   Us_hat[b,o] = sum_i ftd[b,i]*(Cwr+iCwi)[i,o,m]
// FT tiles (32KB) async-staged into LDS. 8 waves -> 8 N-tiles of 16.
// ---------------------------------------------------------------------------
__global__ void __launch_bounds__(256) mode_gemm(
    const float* __restrict__ dr, const float* __restrict__ di,
    const float* __restrict__ xr, const float* __restrict__ xi,
    const float* __restrict__ Awr, const float* __restrict__ Awi,
    const float* __restrict__ Bwr, const float* __restrict__ Bwi,
    const float* __restrict__ Cwr, const float* __restrict__ Cwi,
    float* __restrict__ udr_o, float* __restrict__ udi_o,
    float* __restrict__ usr_o, float* __restrict__ usi_o) {
  __shared__ float s_dr[2048], s_di[2048], s_xr[2048], s_xi[2048];
  const int m = blockIdx.x;
  const int tid = threadIdx.x;
  for (int idx = tid; idx < 2048; idx += 256) {
    int bq = idx >> 7, i = idx & 127;
    size_t g = ((size_t)((bq << 6) + m) << 7) + i;
    async_copy_b32(&s_dr[idx], &dr[g]);
    async_copy_b32(&s_di[idx], &di[g]);
    async_copy_b32(&s_xr[idx], &xr[g]);
    async_copy_b32(&s_xi[idx], &xi[g]);
  }
  async_wait0();
  __syncthreads();

  const int lane = tid & 31;
  const int o0 = (tid >> 5) << 4;
  const int bm = lane & 15;
  const int kh = (lane >> 4) << 1;
  v8f udr = {0.f, 0.f, 0.f, 0.f, 0.f, 0.f, 0.f, 0.f};
  v8f udi = udr, usr = udr, usi = udr;

  for (int k0 = 0; k0 < 128; k0 += 4) {
    const int ka = k0 + kh;
    const int ab = (bm << 7) + ka;
    v2f a_dr = {s_dr[ab], s_dr[ab + 1]};
    v2f a_di = {s_di[ab], s_di[ab + 1]};
    v2f a_xr = {s_xr[ab], s_xr[ab + 1]};
    v2f a_xi = {s_xi[ab], s_xi[ab + 1]};
    v2f n_di = -a_di;
    v2f n_xi = -a_xi;

    const size_t wb0 = ((size_t)ka * 128 + (size_t)(o0 + bm)) * 64 + m;
    const size_t wb1 = wb0 + (size_t)128 * 64;
    // speculative prefetch of the next K-step's weight lines (L2-resident)
    __builtin_prefetch(&Awr[wb0 + (size_t)4 * 128 * 64], 0, 1);
    __builtin_prefetch(&Cwr[wb0 + (size_t)4 * 128 * 64], 0, 1);
    v2f b_ar = {Awr[wb0], Awr[wb1]};
    v2f b_ai = {Awi[wb0], Awi[wb1]};
    v2f b_br = {Bwr[wb0], Bwr[wb1]};
    v2f b_bi = {Bwi[wb0], Bwi[wb1]};
    v2f b_cr = {Cwr[wb0], Cwr[wb1]};
    v2f b_ci = {Cwi[wb0], Cwi[wb1]};

    udr = wmma4(a_dr, b_ar, udr);
    udr = wmma4(n_di, b_ai, udr);
    udr = wmma4(a_xr, b_br, udr);
    udr = wmma4(n_xi, b_bi, udr);
    udi = wmma4(a_dr, b_ai, udi);
    udi = wmma4(a_di, b_ar, udi);
    udi = wmma4(a_xr, b_bi, udi);
    udi = wmma4(a_xi, b_br, udi);
    usr = wmma4(a_dr, b_cr, usr);
    usr = wmma4(n_di, b_ci, usr);
    usi = wmma4(a_dr, b_ci, usi);
    usi = wmma4(a_di, b_cr, usi);
  }

  const int mhi = (lane >> 4) << 3;
#pragma unroll
  for (int r = 0; r < 8; ++r) {
    int bb = r + mhi;
    size_t g = ((size_t)((bb << 6) + m) << 7) + (o0 + bm);
    udr_o[g] = udr[r];
    udi_o[g] = udi[r];
    usr_o[g] = usr[r];
    usi_o[g] = usi[r];
  }
}

// ---------------------------------------------------------------------------
// LayerNorm (over 128 ch) + exact GELU, in place on x of shape (16, 128).
// ---------------------------------------------------------------------------
__global__ void ln_gelu(float* __restrict__ x, const float* __restrict__ w,
                        const float* __restrict__ bias) {
  __shared__ float red[128];
  int b = blockIdx.x, ch = threadIdx.x;
  float v = x[(b << 7) + ch];
  red[ch] = v;
  __syncthreads();
  for (int s = 64; s > 0; s >>= 1) {
    if (ch < s) red[ch] += red[ch + s];
    __syncthreads();
  }
  float mu = red[0] * (1.f / 128.f);
  __syncthreads();
  float d = v - mu;
  red[ch] = d * d;
  __syncthreads();
  for (int s = 64; s > 0; s >>= 1) {
    if (ch < s) red[ch] += red[ch + s];
    __syncthreads();
  }
  float var = red[0] * (1.f / 128.f);
  float y = d * rsqrtf(var + 1e-5f) * w[ch] + bias[ch];
  x[(b << 7) + ch] = 0.5f * y * (1.f + erff(y * 0.70710678118654752f));
}

// ---------------------------------------------------------------------------
// Reconstruction step:
//   xp = x + Us;  xc = concat(xp, Ud) over k;  e = xc@rc_e; o = xc@rc_o
//   out[:,2j] = e; out[:,2j+1] = o
// ---------------------------------------------------------------------------
__global__ void recon(const float* __restrict__ xin, const float* __restrict__ us,
                      const float* __restrict__ ud, const float* __restrict__ rce,
                      const float* __restrict__ rco, float* __restrict__ out,
                      int n) {
  int idx = blockIdx.x * 256 + threadIdx.x;
  if (idx >= (n << 11)) return;
  int k = idx & 7;
  int c = (idx >> 3) & 15;
  int j = (idx >> 7) % n;
  int b = (idx >> 7) / n;
  size_t rowb = (((size_t)b * n + j) << 7) + (c << 3);
  float e = 0.f, o = 0.f;
#pragma unroll
  for (int t = 0; t < 8; ++t) {
    float xp = xin[rowb + t] + us[rowb + t];
    float dd = ud[rowb + t];
    e = fmaf(xp, rce[t * 8 + k], e);
    e = fmaf(dd, rce[(t + 8) * 8 + k], e);
    o = fmaf(xp, rco[t * 8 + k], o);
    o = fmaf(dd, rco[(t + 8) * 8 + k], o);
  }
  size_t ob = ((size_t)((b * (n << 1)) + (j << 1)) * 16 + c) * 8 + k;
  out[ob] = e;
  out[ob + 128] = o;
}

// ---------------------------------------------------------------------------
extern "C" void kernel_launch(void* const* d_in, const int* in_sizes, int n_in,
                              void* d_out, int out_size, void* d_ws,
                              size_t ws_size, hipStream_t stream) {
  (void)in_sizes; (void)n_in; (void)out_size; (void)ws_size;
  const float* x_in = (const float*)d_in[0];
  const float* Awr = (const float*)d_in[1];
  const float* Awi = (const float*)d_in[2];
  const float* Bwr = (const float*)d_in[3];
  const float* Bwi = (const float*)d_in[4];
  const float* Cwr = (const float*)d_in[5];
  const float* Cwi = (const float*)d_in[6];
  const float* ln_w = (const float*)d_in[7];
  const float* ln_b = (const float*)d_in[8];
  const float* ec_s = (const float*)d_in[9];
  const float* ec_d = (const float*)d_in[10];
  const float* rc_e = (const float*)d_in[11];
  const float* rc_o = (const float*)d_in[12];
  float* out = (float*)d_out;
  float* ws = (float*)d_ws;

  constexpr size_t TW = 4096;
  constexpr size_t XB = (size_t)16 * 4096 * 128;
  constexpr size_t FT = (size_t)16 * 64 * 128;
  constexpr size_t LVL = (size_t)16 * 8191 * 128;

  float* twc = ws;
  float* tws = twc + TW;
  float* xb0 = tws + TW;
  float* xb1 = xb0 + XB;
  float* dbuf = xb1 + XB;
  float* ftd_r = dbuf + XB;
  float* ftd_i = ftd_r + FT;
  float* ftx_r = ftd_i + FT;
  float* ftx_i = ftx_r + FT;
  float* udh_r = ftx_i + FT;
  float* udh_i = udh_r + FT;
  float* ush_r = udh_i + FT;
  float* ush_i = ush_r + FT;
  float* Ud_all = ush_i + FT;
  float* Us_all = Ud_all + LVL;

  tw_init<<<16, 256, 0, stream>>>(twc, tws);

  // ---------------- analysis (13 levels, 8192 -> 1) ----------------
  const float* src = x_in;
  for (int i = 0; i < 13; ++i) {
    int n = 8192 >> (i + 1);
    int l = n / 2 + 1;
    if (l > 64) l = 64;
    int mt = (l + 15) >> 4;
    int jt = (n + 15) >> 4;
    float* xd = (i & 1) ? xb1 : xb0;
    size_t off = (size_t)(8192 - (8192 >> i)) * 2048;

    wavelet_fwd<<<8 * n, 256, 0, stream>>>(src, ec_s, ec_d, xd, dbuf, n);
    dft_fwd_wmma<<<dim3(mt, 16), 256, 0, stream>>>(dbuf, ftd_r, ftd_i, twc,
                                                   tws, n);
    dft_fwd_wmma<<<dim3(mt, 16), 256, 0, stream>>>(xd, ftx_r, ftx_i, twc, tws,
                                                   n);
    mode_gemm<<<l, 256, 0, stream>>>(ftd_r, ftd_i, ftx_r, ftx_i, Awr, Awi, Bwr,
                                     Bwi, Cwr, Cwi, udh_r, udh_i, ush_r, ush_i);
    dft_inv_wmma<<<dim3(jt, 16), 256, 0, stream>>>(udh_r, udh_i, Ud_all + off,
                                                   twc, tws, n, l);
    dft_inv_wmma<<<dim3(jt, 16), 256, 0, stream>>>(ush_r, ush_i, Us_all + off,
                                                   twc, tws, n, l);
    src = xd;
  }

  // ---------------- bottleneck: LayerNorm + GELU on (16,1,16,8) -----
  float* bot = (float*)src;  // xb0 (last level index 12 is even)
  ln_gelu<<<16, 128, 0, stream>>>(bot, ln_w, ln_b);

  // ---------------- synthesis (1 -> 8192) --------------------------
  const float* cur = bot;
  for (int i = 12; i >= 0; --i) {
    int n = 8192 >> (i + 1);
    size_t off = (size_t)(8192 - (8192 >> i)) * 2048;
    float* dst = (i == 0) ? out : ((cur == xb0) ? xb1 : xb0);
    recon<<<8 * n, 256, 0, stream>>>(cur, Us_all + off, Ud_all + off, rc_e,
                                     rc_o, dst, n);
    cur = dst;
  }
}